// EuclideanMessagePassing_463856468032
// MI455X (gfx1250) — compile-verified
//
#include <hip/hip_runtime.h>

typedef __attribute__((ext_vector_type(2))) float v2f;
typedef __attribute__((ext_vector_type(8))) float v8f;

#define DFEAT 128           // D_IN == D_OUT == 128
#define NNODES 50000
#define MTILES (NNODES / 16) // 3125
#define NTILES (DFEAT / 16)  // 8

// ---------------------------------------------------------------------------
// Kernel 0: zero the aggregation workspace (harness poisons d_ws with 0xAA).
// ---------------------------------------------------------------------------
__global__ __launch_bounds__(256) void emp_zero_kernel(float4* __restrict__ agg4, int n4) {
    int i = blockIdx.x * blockDim.x + threadIdx.x;
    const int stride = gridDim.x * blockDim.x;
    const float4 z = {0.f, 0.f, 0.f, 0.f};
    for (; i < n4; i += stride) agg4[i] = z;
}

// ---------------------------------------------------------------------------
// Kernel 1: edge scatter. One wave32 per edge: lane l handles 4 contiguous
// features (float4), multiplies by the edge weight, and scatter-adds with
// global_atomic_add_f32. x (25.6 MB) stays resident in the 192 MB L2, so the
// random gather is L2-bound, not HBM-bound.
// ---------------------------------------------------------------------------
__global__ __launch_bounds__(256) void emp_scatter_kernel(
    const float* __restrict__ x,
    const int* __restrict__ edge_index,   // [2, E] (row 0 = src, row 1 = dst)
    const float* __restrict__ edge_weight,
    float* __restrict__ agg,
    int E) {
    const int lane  = threadIdx.x & 31;
    const int wave  = (blockIdx.x * blockDim.x + threadIdx.x) >> 5;
    const int waves = (gridDim.x * blockDim.x) >> 5;
    for (int e = wave; e < E; e += waves) {
        const int   src = edge_index[e];
        const int   dst = edge_index[E + e];
        const float w   = edge_weight[e];
        const float4 v  = ((const float4*)(x + (size_t)src * DFEAT))[lane];
        float* a = agg + (size_t)dst * DFEAT + lane * 4;
        atomicAdd(a + 0, v.x * w);
        atomicAdd(a + 1, v.y * w);
        atomicAdd(a + 2, v.z * w);
        atomicAdd(a + 3, v.w * w);
    }
}

// ---------------------------------------------------------------------------
// Kernel 2: out = relu(agg @ W^T + bias) via V_WMMA_F32_16X16X4_F32.
// One wave computes one 16x16 tile of the output. B fragments (the weights)
// are preloaded into 64 VGPRs and reused for the whole K loop.
//
// f32 WMMA lane layouts (ISA 7.12.2):
//   A 16x4 : lanes 0-15 -> M=lane,   {K=k,   K=k+1} ; lanes 16-31 -> {k+2, k+3}
//   B 4x16 : lanes 0-15 -> N=lane,   {K=k,   K=k+1} ; lanes 16-31 -> {k+2, k+3}
//   C 16x16: lanes 0-15 -> N=lane, VGPR v -> M=v ; lanes 16-31 -> M=v+8
// ---------------------------------------------------------------------------
__global__ __launch_bounds__(256) void emp_gemm_kernel(
    const float* __restrict__ agg,
    const float* __restrict__ weight,  // [D_OUT, D_IN] row-major
    const float* __restrict__ bias,
    float* __restrict__ out) {
    const int lane       = threadIdx.x & 31;
    const int waveInBlk  = threadIdx.x >> 5;          // 0..7
    const int waveGlobal = blockIdx.x * 8 + waveInBlk;
    const int ntile      = waveGlobal & (NTILES - 1); // fixed column tile per wave
    const int n0         = ntile * 16;
    const int mtile0     = waveGlobal >> 3;
    const int mstride    = gridDim.x;                 // (#waves) / NTILES

    const int half = lane >> 4;   // 0: lanes 0-15, 1: lanes 16-31
    const int l15  = lane & 15;

    // Preload B fragments: B[k][n] = W^T[k][n] = weight[n*128 + k]
    v2f bfrag[32];
    const float* wrow = weight + (size_t)(n0 + l15) * DFEAT + half * 2;
#pragma unroll
    for (int kk = 0; kk < 32; ++kk)
        bfrag[kk] = *reinterpret_cast<const v2f*>(wrow + kk * 4);

    const float bval = bias[n0 + l15];

    for (int mt = mtile0; mt < MTILES; mt += mstride) {
        const int m0 = mt * 16;
        const float* arow = agg + (size_t)(m0 + l15) * DFEAT + half * 2;
        v8f c = {};
#pragma unroll
        for (int kk = 0; kk < 32; ++kk) {
            const v2f a = *reinterpret_cast<const v2f*>(arow + kk * 4);
            c = __builtin_amdgcn_wmma_f32_16x16x4_f32(
                    /*neg_a=*/false, a, /*neg_b=*/false, bfrag[kk],
                    /*c_mod=*/(short)0, c, /*reuse_a=*/false, /*reuse_b=*/false);
        }
        // Epilogue: bias + ReLU, scatter per the C/D lane layout.
#pragma unroll
        for (int v = 0; v < 8; ++v) {
            const int m = m0 + v + half * 8;
            float val = c[v] + bval;
            out[(size_t)m * DFEAT + n0 + l15] = val > 0.f ? val : 0.f;
        }
    }
}

// ---------------------------------------------------------------------------
// Launch: inputs in setup_inputs() order:
//   d_in[0]=x [N,128] f32, d_in[1]=edge_index [2,E] int, d_in[2]=edge_weight [E] f32,
//   d_in[3]=weight [128,128] f32, d_in[4]=bias [128] f32.  d_out: [N,128] f32.
// d_ws: agg [N,128] f32 (25.6 MB).
// ---------------------------------------------------------------------------
extern "C" void kernel_launch(void* const* d_in, const int* in_sizes, int n_in,
                              void* d_out, int out_size, void* d_ws, size_t ws_size,
                              hipStream_t stream) {
    const float* x           = (const float*)d_in[0];
    const int*   edge_index  = (const int*)d_in[1];
    const float* edge_weight = (const float*)d_in[2];
    const float* weight      = (const float*)d_in[3];
    const float* bias        = (const float*)d_in[4];
    float*       out         = (float*)d_out;
    float*       agg         = (float*)d_ws;

    const int E = in_sizes[2];                 // 625000
    const int aggFloats = NNODES * DFEAT;      // 6.4M floats

    // 1) zero the accumulator workspace
    emp_zero_kernel<<<2048, 256, 0, stream>>>((float4*)agg, aggFloats / 4);

    // 2) edge scatter (atomic fp32 adds into L2)
    emp_scatter_kernel<<<4096, 256, 0, stream>>>(x, edge_index, edge_weight, agg, E);

    // 3) WMMA GEMM + bias + ReLU; 3125 blocks x 8 waves = one wave per 16x16 tile
    emp_gemm_kernel<<<MTILES, 256, 0, stream>>>(agg, weight, bias, out);
}